// MultiLevelAlignedRoIPooling_31705448579245
// MI455X (gfx1250) — compile-verified
//
#include <hip/hip_runtime.h>
#include <math.h>
#include <stdint.h>

#ifndef __has_builtin
#define __has_builtin(x) 0
#endif

#if __has_builtin(__builtin_amdgcn_global_load_async_to_lds_b128) && \
    __has_builtin(__builtin_amdgcn_s_wait_asynccnt)
#define USE_ASYNC_LDS 1
#else
#define USE_ASYNC_LDS 0
#endif

typedef float f4 __attribute__((ext_vector_type(4)));
typedef int   v4i __attribute__((ext_vector_type(4)));
typedef __attribute__((address_space(1))) v4i* gv4i_t;   // global int4*
typedef __attribute__((address_space(3))) v4i* lv4i_t;   // LDS int4*

#define NPP  7
#define CH   256
#define NBOX 512
#define NGP  49   // 7*7
#define NIT  13   // ceil(49 grid points / 4 groups)

struct GPInfo {
  const float* a00; const float* a01; const float* a10; const float* a11;
  float w00, w01, w10, w11;
  float* dst;
  bool valid;
};

__global__ __launch_bounds__(256)
void roi_align_ml_kernel(const float* __restrict__ f2, const float* __restrict__ f3,
                         const float* __restrict__ f4l, const float* __restrict__ f5,
                         const float* __restrict__ boxes, float* __restrict__ out)
{
#if USE_ASYNC_LDS
  __shared__ f4 lbuf[2][4][256];   // [double-buffer][corner][thread] : 32 KB
#endif
  const int tid = threadIdx.x;
  const int blk = blockIdx.x;      // = b*NBOX + n
  const int b   = blk >> 9;

  // ---- per-box parameters (uniform across block) ----
  const float* bp = boxes + (size_t)blk * 4;
  const float by0 = bp[0], bx0 = bp[1], by1 = bp[2], bx1 = bp[3];
  const float bh = by1 - by0, bw = bx1 - bx0;
  const float asq = sqrtf(bh * bw);
  int lv = (int)floorf(log2f(asq * (1.0f / 224.0f))) + 4;
  lv = lv < 2 ? 2 : (lv > 5 ? 5 : lv);
  const int   li     = lv - 2;
  const float inv2lv = 1.0f / (float)(1 << lv);
  const int   Hl     = 256 >> li;                 // level dims (== Wl)
  const float bnd    = (float)(Hl - 1);           // H0/stride - 1
  const float ry0 = by0 * inv2lv - 0.5f;
  const float rx0 = bx0 * inv2lv - 0.5f;
  const float sy  = bh * inv2lv * (1.0f / NPP);
  const float sx  = bw * inv2lv * (1.0f / NPP);
  const float* lp = (li == 0) ? f2 : (li == 1) ? f3 : (li == 2) ? f4l : f5;
  lp += (size_t)b * Hl * Hl * CH;

  const int grp = tid >> 6;          // 4 groups of 64 threads
  const int c4  = (tid & 63) * 4;    // float4 channel quad
  float* outb = out + (size_t)blk * (NGP * CH) + c4;

  auto make = [&](int it) -> GPInfo {
    GPInfo g;
    int gp  = it * 4 + grp;
    g.valid = gp < NGP;
    int gc  = g.valid ? gp : NGP - 1;     // clamp tail; keeps EXEC full
    int py  = gc / NPP, px = gc - py * NPP;
    float gy  = fminf(ry0 + (float)py * sy, bnd);
    float gx  = fminf(rx0 + (float)px * sx, bnd);
    float y0f = floorf(gy), x0f = floorf(gx);
    float ly  = gy - y0f,   lx  = gx - x0f;
    float hy  = 1.0f - ly,  hx  = 1.0f - lx;
    int y0 = (int)y0f, x0i = (int)x0f;
    // gy==bnd  =>  ly==0, so clamping the +1 index matches the reference's
    // zero-padded gather exactly (0-weight corner).
    int y1 = (y0  + 1 < Hl) ? y0  + 1 : Hl - 1;
    int x1 = (x0i + 1 < Hl) ? x0i + 1 : Hl - 1;
    const float* r0 = lp + (size_t)y0 * Hl * CH;
    const float* r1 = lp + (size_t)y1 * Hl * CH;
    g.a00 = r0 + x0i * CH + c4;
    g.a01 = r0 + x1  * CH + c4;
    g.a10 = r1 + x0i * CH + c4;
    g.a11 = r1 + x1  * CH + c4;
    g.w00 = hy * hx; g.w01 = hy * lx; g.w10 = ly * hx; g.w11 = ly * lx;
    g.dst = outb + (size_t)gc * CH;
    return g;
  };

#if USE_ASYNC_LDS
  auto issue = [&](const GPInfo& g, int buf) {
    // Generic->LDS offset is architecturally addr[31:0] (ISA aperture rules);
    // build properly typed AS(1)/AS(3) int4* via integer casts.
    unsigned l0 = (unsigned)(uintptr_t)&lbuf[buf][0][tid];
    unsigned l1 = (unsigned)(uintptr_t)&lbuf[buf][1][tid];
    unsigned l2 = (unsigned)(uintptr_t)&lbuf[buf][2][tid];
    unsigned l3 = (unsigned)(uintptr_t)&lbuf[buf][3][tid];
    __builtin_amdgcn_global_load_async_to_lds_b128((gv4i_t)(uintptr_t)g.a00, (lv4i_t)(uintptr_t)l0, 0, 0);
    __builtin_amdgcn_global_load_async_to_lds_b128((gv4i_t)(uintptr_t)g.a01, (lv4i_t)(uintptr_t)l1, 0, 0);
    __builtin_amdgcn_global_load_async_to_lds_b128((gv4i_t)(uintptr_t)g.a10, (lv4i_t)(uintptr_t)l2, 0, 0);
    __builtin_amdgcn_global_load_async_to_lds_b128((gv4i_t)(uintptr_t)g.a11, (lv4i_t)(uintptr_t)l3, 0, 0);
    __asm__ __volatile__("" ::: "memory");
  };
#endif

  GPInfo cur = make(0);
#if USE_ASYNC_LDS
  issue(cur, 0);
#endif

  for (int it = 0; it < NIT; ++it) {
    GPInfo nxt = cur;
    const bool has_next = (it + 1) < NIT;
    if (has_next) {
      nxt = make(it + 1);
#if USE_ASYNC_LDS
      issue(nxt, (it + 1) & 1);     // prefetch next grid point's corners
#endif
    }
    f4 v00, v01, v10, v11;
#if USE_ASYNC_LDS
    // 4 async loads per iteration per wave; async loads complete in order,
    // so cnt<=4 guarantees the current buffer's 4 corners have landed.
    if (has_next) __builtin_amdgcn_s_wait_asynccnt(4);
    else          __builtin_amdgcn_s_wait_asynccnt(0);
    __asm__ __volatile__("" ::: "memory");
    const int buf = it & 1;
    v00 = lbuf[buf][0][tid];
    v01 = lbuf[buf][1][tid];
    v10 = lbuf[buf][2][tid];
    v11 = lbuf[buf][3][tid];
#else
    if (has_next) {
      __builtin_prefetch(nxt.a00, 0, 1);
      __builtin_prefetch(nxt.a01, 0, 1);
      __builtin_prefetch(nxt.a10, 0, 1);
      __builtin_prefetch(nxt.a11, 0, 1);
    }
    v00 = *(const f4*)cur.a00;
    v01 = *(const f4*)cur.a01;
    v10 = *(const f4*)cur.a10;
    v11 = *(const f4*)cur.a11;
#endif
    f4 acc = cur.w00 * v00 + cur.w01 * v01 + cur.w10 * v10 + cur.w11 * v11;
    if (cur.valid) {
      // Output is never re-read: stream past L2 (NT), keep L2 for features.
      __builtin_nontemporal_store(acc, (f4*)cur.dst);
    }
    cur = nxt;
  }
}

extern "C" void kernel_launch(void* const* d_in, const int* in_sizes, int n_in,
                              void* d_out, int out_size, void* d_ws, size_t ws_size,
                              hipStream_t stream) {
  const float* f2    = (const float*)d_in[0];
  const float* f3    = (const float*)d_in[1];
  const float* f4l   = (const float*)d_in[2];
  const float* f5    = (const float*)d_in[3];
  const float* boxes = (const float*)d_in[4];
  float* out = (float*)d_out;
  dim3 grid(2 * NBOX);   // one block per (b, n) box
  roi_align_ml_kernel<<<grid, 256, 0, stream>>>(f2, f3, f4l, f5, boxes, out);
}